// ODEG_39187281608729
// MI455X (gfx1250) — compile-verified
//
#include <hip/hip_runtime.h>
#include <hip/hip_bf16.h>

typedef float v2f __attribute__((ext_vector_type(2)));
typedef float v8f __attribute__((ext_vector_type(8)));

#define B_ 16
#define N_ 307
#define T_ 12
#define C_ 64
#define BN 4912            // B_*N_
#define TC 768             // T_*C_
#define BNT 58944          // B_*N_*T_
#define BIG 3772416        // B_*N_*T_*C_   (compact)
#define NPAD 320
#define BIGP 3932160       // B_*NPAD*T_*C_ (node-padded)

__device__ __forceinline__ float sigf(float x) { return 1.0f / (1.0f + __expf(-x)); }

// ---------------------------------------------------------------------------
// Batched fp32 WMMA GEMM, guard-free inner loop:
//   C[0:Mstore, 0:N] = scale * (A @ B) + bias[col]
// Requirements (enforced by padded operands at call sites):
//   M % 16 == 0, M >= 32, N % 16 == 0, K % 64 == 0
//   A has M x K valid elements (lda), B has K x N valid (ldb).
// Block: 256 threads = 8 waves as 2(M) x 4(N) 16x16 tiles (32x64 per block).
// A block is LDS-staged (coalesced float4 loads, ds_load_b64 fragments);
// B fragments are coalesced 64B row segments from global.
// ---------------------------------------------------------------------------
__global__ void ODEG_gemm_wmma(const float* __restrict__ Am, const float* __restrict__ Bm,
                               float* __restrict__ Cm, const float* __restrict__ bias,
                               int M, int N, int K, int lda, int ldb, int ldc, int Mstore,
                               long sA, long sB, long sC, float scale)
{
    __shared__ float ldsA[32 * 64];
    Am += (long)blockIdx.y * sA;
    Bm += (long)blockIdx.y * sB;
    Cm += (long)blockIdx.y * sC;

    int tilesN = N >> 4;
    int gN = (tilesN + 3) >> 2;            // N tile-groups of 4
    int gm = blockIdx.x / gN;
    int gn = blockIdx.x % gN;
    int bm = gm * 32;  if (bm > M - 32) bm = M - 32;
    int bn = gn * 64;

    int wave = threadIdx.x >> 5;
    int wm = wave >> 2;                    // 0..1
    int wn = wave & 3;                     // 0..3
    int lane = threadIdx.x & 31;
    int l16 = lane & 15;
    int half = lane >> 4;

    int tm = bm + wm * 16;
    int tn = bn + wn * 16;  if (tn > N - 16) tn = N - 16;   // clamp (dup work ok)
    int bcol = tn + l16;

    // cooperative A-load mapping: 32 rows x 64 cols, 8 floats per thread
    int ar = threadIdx.x >> 3;             // 0..31
    int ac = (threadIdx.x & 7) * 8;        // 0..56

    v8f acc = {};
    for (int kc0 = 0; kc0 < K; kc0 += 64) {
        __syncthreads();
        {
            const float* src = Am + (long)(bm + ar) * lda + kc0 + ac;
            *(float4*)&ldsA[ar * 64 + ac]     = *(const float4*)src;
            *(float4*)&ldsA[ar * 64 + ac + 4] = *(const float4*)(src + 4);
        }
        __syncthreads();
        const float* brow = Bm + (long)kc0 * ldb + bcol;
#pragma unroll
        for (int kk = 0; kk < 64; kk += 4) {
            int ka = kk + half * 2;
            v2f af = *(const v2f*)&ldsA[(wm * 16 + l16) * 64 + ka];
            v2f bf;
            bf.x = brow[(long)(ka)     * ldb];
            bf.y = brow[(long)(ka + 1) * ldb];
            acc = __builtin_amdgcn_wmma_f32_16x16x4_f32(false, af, false, bf,
                                                        (short)0, acc, false, false);
        }
    }

    float bv = bias ? bias[bcol] : 0.0f;
#pragma unroll
    for (int r = 0; r < 8; ++r) {
        int row = tm + r + half * 8;
        if (row < Mstore) Cm[(long)row * ldc + bcol] = acc[r] * scale + bv;
    }
}

// ---------------------------------------------------------------------------
// padded copies
// ---------------------------------------------------------------------------
__global__ void ODEG_asp(const float* __restrict__ adj, const float* __restrict__ mask,
                         float* __restrict__ ASP)
{
    int idx = blockIdx.x * blockDim.x + threadIdx.x;
    if (idx >= NPAD * NPAD) return;
    int i = idx / NPAD, j = idx % NPAD;
    ASP[idx] = (i < N_ && j < N_) ? adj[i * N_ + j] + mask[i * N_ + j] : 0.0f;
}

// XP[b][320][768] = x zero-padded over node dim
__global__ void ODEG_xpad(const float* __restrict__ x, float* __restrict__ XP)
{
    long idx = (long)blockIdx.x * blockDim.x + threadIdx.x;
    if (idx >= (long)BIGP) return;
    int tc = (int)(idx % TC);
    long bn = idx / TC;
    int n = (int)(bn % NPAD);
    int b = (int)(bn / NPAD);
    XP[idx] = (n < N_) ? x[((long)b * N_ + n) * TC + tc] : 0.0f;
}

// EMP[b][320][16] = EM zero-padded (rows and cols)
__global__ void ODEG_empad(const float* __restrict__ EM, float* __restrict__ EMP)
{
    int idx = blockIdx.x * blockDim.x + threadIdx.x;
    if (idx >= B_ * NPAD * 16) return;
    int t = idx & 15;
    int n = (idx >> 4) % NPAD;
    int b = idx / (NPAD * 16);
    EMP[idx] = (n < N_ && t < 12) ? EM[((long)b * N_ + n) * 12 + t] : 0.0f;
}

// ---------------------------------------------------------------------------
// W_g = (1+beta)*W0 - beta*(W0 W0^T W0),  W0 = (w*clip(d,0,1)) @ w^T ; 5 blocks
// ---------------------------------------------------------------------------
__global__ void ODEG_wprep(const float* __restrict__ wn, const float* __restrict__ dn,
                           const float* __restrict__ w3, const float* __restrict__ d3,
                           float* __restrict__ W5)
{
    int g = blockIdx.x;
    const float* w = (g == 0) ? wn : w3 + (g - 1) * 4096;
    const float* d = (g == 0) ? dn : d3 + (g - 1) * 64;
    __shared__ float W[4096];
    __shared__ float Tm[4096];
    int tid = threadIdx.x;
    for (int e = tid; e < 4096; e += 256) {
        int i = e >> 6, k = e & 63;
        float s = 0.0f;
        for (int j = 0; j < 64; ++j) {
            float dd = fminf(fmaxf(d[j], 0.0f), 1.0f);
            s += w[i * 64 + j] * dd * w[k * 64 + j];
        }
        W[e] = s;
    }
    __syncthreads();
    for (int e = tid; e < 4096; e += 256) {
        int i = e >> 6, k = e & 63;
        float s = 0.0f;
        for (int j = 0; j < 64; ++j) s += W[i * 64 + j] * W[k * 64 + j];
        Tm[e] = s;
    }
    __syncthreads();
    for (int e = tid; e < 4096; e += 256) {
        int i = e >> 6, k = e & 63;
        float s = 0.0f;
        for (int j = 0; j < 64; ++j) s += Tm[i * 64 + j] * W[j * 64 + k];
        W5[g * 4096 + e] = 1.6f * W[e] - 0.6f * s;   // beta = 0.6
    }
}

// ---------------------------------------------------------------------------
// node temporal attention
// ---------------------------------------------------------------------------
__global__ void ODEG_alar(const float* __restrict__ x, const float* __restrict__ w3,
                          const float* __restrict__ w4, float* __restrict__ AL,
                          float* __restrict__ AR)
{
    long idx = (long)blockIdx.x * blockDim.x + threadIdx.x;
    if (idx >= (long)BIG) return;
    int c = idx & 63;
    int o = (idx >> 6) % 12;
    long bn = idx / 768;
    const float* xb = x + bn * 768 + c;
    float sa = 0.0f, sr = 0.0f;
    for (int t = 0; t < 12; ++t) {
        float xv = xb[t * 64];
        sa += xv * w3[t * 12 + o];
        sr += xv * w4[t * 12 + o];
    }
    AL[idx] = sa;
    AR[idx] = sr;
}

__global__ void ODEG_attn_node(const float* __restrict__ AL, const float* __restrict__ AR,
                               float* __restrict__ ATTN)
{
    int idx = blockIdx.x * blockDim.x + threadIdx.x;
    if (idx >= B_ * 64 * 144) return;
    int p = idx % 12;
    int o = (idx / 12) % 12;
    int c = (idx / 144) % 64;
    int b = idx / 9216;
    const float* al = AL + ((long)b * N_ * 12 + o) * 64 + c;
    const float* ar = AR + ((long)b * N_ * 12 + p) * 64 + c;
    float s = 0.0f;
    for (int n = 0; n < N_; ++n) s += al[(long)n * 768] * ar[(long)n * 768];
    ATTN[idx] = sigf(s);
}

__global__ void ODEG_x0n(const float* __restrict__ x, const float* __restrict__ XSP,
                         const float* __restrict__ XW, const float* __restrict__ ATTN,
                         const float* __restrict__ alpha, float* __restrict__ X0N)
{
    long idx = (long)blockIdx.x * blockDim.x + threadIdx.x;
    if (idx >= (long)BIG) return;
    int c = idx & 63;
    int o = (idx >> 6) % 12;
    long bn = idx / 768;
    int n = (int)(bn % N_);
    int b = (int)(bn / N_);
    const float* at = ATTN + ((long)(b * 64 + c) * 12 + o) * 12;
    const float* xp = x + bn * 768 + c;
    float tn = 0.0f;
    for (int p = 0; p < 12; ++p) tn += at[p] * xp[p * 64];
    float a = sigf(alpha[n]);
    X0N[idx] = 0.5f * a * XSP[idx] + tn + XW[idx] - x[idx];
}

// ---------------------------------------------------------------------------
// Fused MHSA with node-padded Q/K ([b][320][768]): guard-free WMMA loop.
// ---------------------------------------------------------------------------
__global__ void ODEG_mhsa(const float* __restrict__ Q, const float* __restrict__ Km,
                          const float* __restrict__ V, float* __restrict__ O)
{
    __shared__ float s[16 * NPAD];
    int rt = blockIdx.x;                 // row tile 0..19
    int h = blockIdx.y;                  // head 0..31
    int b = blockIdx.z;                  // batch 0..15
    int lane = threadIdx.x & 31;
    int l16 = lane & 15;
    int half = lane >> 4;
    const float* Qp = Q + (long)b * NPAD * 768 + h * 24;
    const float* Kp = Km + (long)b * NPAD * 768 + h * 24;
    const float* Vp = V + (long)b * N_ * 256 + h * 8;
    float* Op = O + (long)b * N_ * 256 + h * 8;
    int qrow = rt * 16 + l16;
    const float norm = 0.2041241452319315f;   // 1/sqrt(24)
    for (int mt = 0; mt < 20; ++mt) {
        v8f acc = {};
        int mcol = mt * 16 + l16;
        const float* qr = Qp + (long)qrow * 768;
        const float* kr = Kp + (long)mcol * 768;
#pragma unroll
        for (int k0 = 0; k0 < 24; k0 += 4) {
            int ka = k0 + half * 2;
            v2f af = *(const v2f*)&qr[ka];
            v2f bf = *(const v2f*)&kr[ka];
            acc = __builtin_amdgcn_wmma_f32_16x16x4_f32(false, af, false, bf,
                                                        (short)0, acc, false, false);
        }
#pragma unroll
        for (int r = 0; r < 8; ++r) {
            int row = r + half * 8;
            int m = mt * 16 + l16;
            s[row * NPAD + m] = (m < N_) ? acc[r] * norm : -1e30f;
        }
    }
    __syncthreads();
    if (lane < 16) {
        float mx = -1e30f;
        for (int m = 0; m < N_; ++m) mx = fmaxf(mx, s[lane * NPAD + m]);
        float sum = 0.0f;
        for (int m = 0; m < N_; ++m) {
            float e = __expf(s[lane * NPAD + m] - mx);
            s[lane * NPAD + m] = e;
            sum += e;
        }
        float inv = 1.0f / sum;
        for (int m = 0; m < N_; ++m) s[lane * NPAD + m] *= inv;
    }
    __syncthreads();
    for (int t = 0; t < 4; ++t) {
        int idx = lane + 32 * t;
        int row = idx >> 3, dv = idx & 7;
        float a = 0.0f;
        for (int m = 0; m < N_; ++m) a += s[row * NPAD + m] * Vp[(long)m * 256 + dv];
        int grow = rt * 16 + row;
        if (grow < N_) Op[(long)grow * 256 + dv] = a;
    }
}

// ---------------------------------------------------------------------------
// block3 helpers (node-padded Z layout [b][320][64], pad rows stay zero)
// ---------------------------------------------------------------------------
__global__ void ODEG_zinit(const float* __restrict__ X1, float* __restrict__ Z0,
                           float* __restrict__ Z, int g)
{
    int idx = blockIdx.x * blockDim.x + threadIdx.x;
    if (idx >= B_ * NPAD * 64) return;
    int c = idx & 63;
    int n = (idx >> 6) % NPAD;
    int b = idx / (NPAD * 64);
    float v = (n < N_) ? X1[((long)b * N_ + n) * 256 + g * 64 + c] : 0.0f;
    Z0[idx] = v;
    Z[idx] = v;
}

__global__ void ODEG_sumsq(const float* __restrict__ Z, float* __restrict__ SS)
{
    int b = blockIdx.x, c = threadIdx.x;
    float s = 0.0f;
    const float* p = Z + (long)b * NPAD * 64 + c;
    for (int n = 0; n < NPAD; ++n) {    // pad rows are zero
        float v = p[(long)n * 64];
        s += v * v;
    }
    SS[b * 64 + c] = s;
}

__global__ void ODEG_zupdate(float* __restrict__ Z, const float* __restrict__ ZSP,
                             const float* __restrict__ ZW, const float* __restrict__ Z0,
                             const float* __restrict__ SS, const float* __restrict__ alpha3,
                             const float* __restrict__ w31p, const float* __restrict__ w41p,
                             float* __restrict__ X1S, int g, int step)
{
    int idx = blockIdx.x * blockDim.x + threadIdx.x;
    if (idx >= B_ * NPAD * 64) return;
    int c = idx & 63;
    int n = (idx >> 6) % NPAD;
    int b = idx / (NPAD * 64);
    if (n >= N_) return;                 // pad rows stay zero
    float attc = sigf(w31p[0] * w41p[0] * SS[b * 64 + c]);
    float a = sigf(alpha3[g * N_ + n]);
    float zv = Z[idx];
    float zn = 0.5f * a * ZSP[idx] + attc * zv + ZW[idx] + Z0[idx] - 2.0f * zv;
    Z[idx] = zn;
    X1S[(((long)b * N_ + n) * 12 + (step * 4 + g)) * 64 + c] = zn;
}

// ---------------------------------------------------------------------------
// Rank-collapsed edge branch (edge0[b,n,m,t] = em[b,n,t]+em[b,m,t])
// ---------------------------------------------------------------------------
__global__ void ODEG_em(const float* __restrict__ x, float* __restrict__ EM)
{
    int idx = blockIdx.x * blockDim.x + threadIdx.x;
    if (idx >= BNT) return;
    const float* p = x + (long)idx * 64;
    float s = 0.0f;
    for (int c = 0; c < 64; ++c) s += p[c];
    EM[idx] = s * (1.0f / 64.0f);
}

__global__ void ODEG_eoep(const float* __restrict__ EM, const float* __restrict__ w3,
                          const float* __restrict__ w4, float* __restrict__ EO,
                          float* __restrict__ EP)
{
    int idx = blockIdx.x * blockDim.x + threadIdx.x;
    if (idx >= BNT) return;
    int o = idx % 12;
    long bn = idx / 12;
    const float* e = EM + bn * 12;
    float so = 0.0f, sp = 0.0f;
    for (int t = 0; t < 12; ++t) {
        so += e[t] * w3[t * 12 + o];
        sp += e[t] * w4[t * 12 + o];
    }
    EO[idx] = so;
    EP[idx] = sp;
}

__global__ void ODEG_esums(const float* __restrict__ EO, const float* __restrict__ EP,
                           float* __restrict__ SO, float* __restrict__ SP,
                           float* __restrict__ CR)
{
    int b = blockIdx.x;
    int o = threadIdx.x / 12, p = threadIdx.x % 12;
    const float* eo = EO + (long)b * N_ * 12;
    const float* ep = EP + (long)b * N_ * 12;
    float s = 0.0f;
    for (int n = 0; n < N_; ++n) s += eo[n * 12 + o] * ep[n * 12 + p];
    CR[b * 144 + threadIdx.x] = s;
    if (p == 0) {
        float t = 0.0f;
        for (int n = 0; n < N_; ++n) t += eo[n * 12 + o];
        SO[b * 12 + o] = t;
    }
    if (o == 0) {
        float t = 0.0f;
        for (int n = 0; n < N_; ++n) t += ep[n * 12 + p];
        SP[b * 12 + p] = t;
    }
}

__global__ void ODEG_atte(const float* __restrict__ EO, const float* __restrict__ EP,
                          const float* __restrict__ SO, const float* __restrict__ SP,
                          const float* __restrict__ CR, float* __restrict__ ATTE)
{
    int idx = blockIdx.x * blockDim.x + threadIdx.x;
    if (idx >= BN * 144) return;
    int p = idx % 12;
    int o = (idx / 12) % 12;
    long bn = idx / 144;
    int b = (int)(bn / N_);
    float eo = EO[bn * 12 + o], ep = EP[bn * 12 + p];
    float v = 307.0f * eo * ep + eo * SP[b * 12 + p] + SO[b * 12 + o] * ep + CR[b * 144 + o * 12 + p];
    ATTE[idx] = sigf(v);
}

__global__ void ODEG_v1e(const float* __restrict__ EM, const float* __restrict__ ATTE,
                         float* __restrict__ V1E)
{
    int idx = blockIdx.x * blockDim.x + threadIdx.x;
    if (idx >= BNT) return;
    int p = idx % 12;
    long bn = idx / 12;
    const float* e = EM + bn * 12;
    const float* a = ATTE + bn * 144 + p;
    float s = 0.0f;
    for (int t = 0; t < 12; ++t) s += e[t] * a[t * 12];
    V1E[idx] = s;
}

__global__ void ODEG_v1t(const float* __restrict__ fce_w, const float* __restrict__ V1E,
                         float* __restrict__ V1T)
{
    int idx = blockIdx.x * blockDim.x + threadIdx.x;
    if (idx >= B_ * 12 * 64) return;
    int c = idx & 63;
    int p = (idx >> 6) % 12;
    int b = idx / 768;
    float s = 0.0f;
    for (int n = 0; n < N_; ++n)
        s += fce_w[n * 64 + c] * V1E[((long)b * N_ + n) * 12 + p];
    V1T[idx] = s;
}

__global__ void ODEG_m2(const float* __restrict__ fce_w, const float* __restrict__ ATTE,
                        float* __restrict__ M2)
{
    int idx = blockIdx.x * blockDim.x + threadIdx.x;
    if (idx >= B_ * 144 * 64) return;
    int c = idx & 63;
    int p = (idx >> 6) % 12;
    int t = (idx >> 6) / 12 % 12;
    int b = idx / (144 * 64);
    float s = 0.0f;
    for (int n = 0; n < N_; ++n)
        s += fce_w[n * 64 + c] * ATTE[((long)b * N_ + n) * 144 + t * 12 + p];
    M2[idx] = s;
}

__global__ void ODEG_hprep(const float* __restrict__ alpha_e, const float* __restrict__ fce_w,
                           const float* __restrict__ ASP, float* __restrict__ H,
                           float* __restrict__ RS)
{
    int idx = blockIdx.x * blockDim.x + threadIdx.x;
    if (idx >= N_ * 64) return;
    int i = idx >> 6, c = idx & 63;
    H[idx] = 0.5f * sigf(alpha_e[i]) * fce_w[idx];
    if (c == 0) {
        float s = 0.0f;
        for (int j = 0; j < N_; ++j) s += ASP[i * NPAD + j];
        RS[i] = s;
    }
}

// AEM pitch is 16 (padded edge GEMM output)
__global__ void ODEG_q1(const float* __restrict__ H, const float* __restrict__ AEM,
                        float* __restrict__ Q1)
{
    int idx = blockIdx.x * blockDim.x + threadIdx.x;
    if (idx >= B_ * 12 * 64) return;
    int c = idx & 63;
    int t = (idx >> 6) % 12;
    int b = idx / 768;
    float s = 0.0f;
    for (int i = 0; i < N_; ++i)
        s += H[i * 64 + c] * AEM[((long)b * N_ + i) * 16 + t];
    Q1[idx] = s;
}

__global__ void ODEG_q2(const float* __restrict__ H, const float* __restrict__ RS,
                        float* __restrict__ Q2)
{
    int c = threadIdx.x;
    if (c >= 64) return;
    float s = 0.0f;
    for (int i = 0; i < N_; ++i) s += H[i * 64 + c] * RS[i];
    Q2[c] = s;
}

__global__ void ODEG_xe(const float* __restrict__ Q1, const float* __restrict__ Q2,
                        const float* __restrict__ V1T, const float* __restrict__ M2,
                        const float* __restrict__ EM, const float* __restrict__ fce_b,
                        float* __restrict__ XE)
{
    long idx = (long)blockIdx.x * blockDim.x + threadIdx.x;
    if (idx >= (long)BIG) return;
    int c = idx & 63;
    int t = (idx >> 6) % 12;
    long bm = idx / 768;
    int b = (int)(bm / N_);
    const float* e = EM + bm * 12;
    float s = Q1[(b * 12 + t) * 64 + c] + Q2[c] * e[t] + V1T[(b * 12 + t) * 64 + c] + fce_b[c];
    for (int tt = 0; tt < 12; ++tt)
        s += e[tt] * M2[(((long)b * 12 + tt) * 12 + t) * 64 + c];
    XE[idx] = s;
}

// ---------------------------------------------------------------------------
// clip + final combines
// ---------------------------------------------------------------------------
__global__ void ODEG_clip(const float* __restrict__ X1S, const float* __restrict__ X0N,
                          const float* __restrict__ clipv, float* __restrict__ X1C)
{
    long idx = (long)blockIdx.x * blockDim.x + threadIdx.x;
    if (idx >= (long)BIG) return;
    float cv = clipv[0];
    float hi = X0N[idx] + cv, lo = X0N[idx] - cv;
    X1C[idx] = fmaxf(lo, fminf(hi, X1S[idx]));
}

__global__ void ODEG_zc(const float* __restrict__ XG, const float* __restrict__ XL,
                        const float* __restrict__ XE, const float* __restrict__ res,
                        float* __restrict__ ZC)
{
    long idx = (long)blockIdx.x * blockDim.x + threadIdx.x;
    if (idx >= (long)BIG) return;
    float xg = XG[idx], xl = XL[idx], xe = XE[idx];
    float sg = sigf(xg), sl = sigf(xl), se = sigf(xe);
    ZC[idx] = (xg * (sl + se) + xl * (se + sg) + xe * (sl + sg) + res[idx]) * (1.0f / 6.0f);
}

__global__ void ODEG_final(const float* __restrict__ RZ1, const float* __restrict__ RZ2,
                           float* __restrict__ out)
{
    long idx = (long)blockIdx.x * blockDim.x + threadIdx.x;
    if (idx >= (long)BIG) return;
    float v = sigf(RZ1[idx]) + RZ2[idx];
    out[idx] = fmaxf(v, 0.0f);
}

// ---------------------------------------------------------------------------
extern "C" void kernel_launch(void* const* d_in, const int* in_sizes, int n_in,
                              void* d_out, int out_size, void* d_ws, size_t ws_size,
                              hipStream_t stream)
{
    const float* x        = (const float*)d_in[0];
    const float* adj      = (const float*)d_in[1];
    const float* mask     = (const float*)d_in[2];
    const float* w3_12    = (const float*)d_in[3];
    const float* w4_12    = (const float*)d_in[4];
    const float* w3_1     = (const float*)d_in[5];
    const float* w4_1     = (const float*)d_in[6];
    const float* alpha_n  = (const float*)d_in[7];
    const float* w_n11    = (const float*)d_in[8];
    const float* d_n11    = (const float*)d_in[9];
    const float* alpha_e  = (const float*)d_in[10];
    const float* alpha_3  = (const float*)d_in[11];
    const float* w_3      = (const float*)d_in[12];
    const float* d_3      = (const float*)d_in[13];
    const float* wq       = (const float*)d_in[14];
    const float* bq       = (const float*)d_in[15];
    const float* wk       = (const float*)d_in[16];
    const float* bk       = (const float*)d_in[17];
    const float* wv       = (const float*)d_in[18];
    const float* bv       = (const float*)d_in[19];
    const float* fc1_w    = (const float*)d_in[20];
    const float* fc1_b    = (const float*)d_in[21];
    const float* fc2_w    = (const float*)d_in[22];
    const float* fc2_b    = (const float*)d_in[23];
    const float* fce_w    = (const float*)d_in[24];
    const float* fce_b    = (const float*)d_in[25];
    const float* clipv    = (const float*)d_in[26];
    float* out = (float*)d_out;

    // -------- workspace layout (floats) --------
    float* W = (float*)d_ws;
    size_t o = 0;
    float* ASP  = W + o; o += (size_t)NPAD * NPAD;
    float* XP   = W + o; o += (size_t)BIGP;                 // padded x
    float* EM   = W + o; o += BNT;
    float* EMP  = W + o; o += (size_t)B_ * NPAD * 16;
    float* EO   = W + o; o += BNT;
    float* EP   = W + o; o += BNT;
    float* SO   = W + o; o += B_ * 12;
    float* SPb  = W + o; o += B_ * 12;
    float* CR   = W + o; o += B_ * 144;
    float* ATTE = W + o; o += (size_t)BN * 144;
    float* V1E  = W + o; o += BNT;
    float* V1T  = W + o; o += B_ * 12 * 64;
    float* M2   = W + o; o += B_ * 144 * 64;
    float* AEM  = W + o; o += (size_t)BN * 16;              // pitch 16
    float* H    = W + o; o += N_ * 64;
    float* RS   = W + o; o += NPAD;
    float* Q1   = W + o; o += B_ * 12 * 64;
    float* Q2   = W + o; o += 64;
    float* W5   = W + o; o += 5 * 4096;
    float* ATTN = W + o; o += B_ * 64 * 144;
    float* SS   = W + o; o += B_ * 64;
    float* VB   = W + o; o += (size_t)BN * 256;
    float* X1   = W + o; o += (size_t)BN * 256;
    float* ZP   = W + o; o += (size_t)B_ * NPAD * 64;
    float* Z0P  = W + o; o += (size_t)B_ * NPAD * 64;
    float* ZSPP = W + o; o += (size_t)B_ * NPAD * 64;
    float* ZWP  = W + o; o += (size_t)B_ * NPAD * 64;
    float* S0 = W + o; o += BIGP;   // AL -> Qpad -> XG
    float* S1 = W + o; o += BIGP;   // AR -> Kpad -> XL
    float* S2 = W + o; o += BIGP;   // Xsp -> X1S -> RZ2
    float* S3 = W + o; o += BIGP;   // XW -> X1C -> RZ1
    float* S4 = W + o; o += BIGP;   // x0n
    float* S5 = W + o; o += BIGP;   // XE
    float* S6 = W + o; o += BIGP;   // Zc
    (void)ws_size; (void)in_sizes; (void)n_in; (void)out_size;

    // gemm: M%16==0 (>=32), N%16==0, K%64==0; stores rows < Mstore
    auto gemm = [&](const float* A, const float* Bm, float* Cm, const float* bias,
                    int M, int N, int K, int lda, int ldb, int ldc, int Mstore,
                    long sA, long sB, long sC, int batch, float scale) {
        int gM = ((M >> 4) + 1) >> 1;
        int gN = ((N >> 4) + 3) >> 2;
        dim3 g(gM * gN, batch, 1);
        ODEG_gemm_wmma<<<g, 256, 0, stream>>>(A, Bm, Cm, bias, M, N, K, lda, ldb, ldc,
                                              Mstore, sA, sB, sC, scale);
    };
    auto gr = [](long n) { return (int)((n + 255) / 256); };

    // ---- prep ----
    ODEG_asp<<<gr(NPAD * NPAD), 256, 0, stream>>>(adj, mask, ASP);
    ODEG_xpad<<<gr(BIGP), 256, 0, stream>>>(x, XP);
    ODEG_wprep<<<5, 256, 0, stream>>>(w_n11, d_n11, w_3, d_3, W5);

    // ---- node ODE (x0n) ----
    ODEG_alar<<<gr(BIG), 256, 0, stream>>>(x, w3_12, w4_12, S0, S1);
    ODEG_attn_node<<<gr(B_ * 64 * 144), 256, 0, stream>>>(S0, S1, ATTN);
    // spatial(x): A_sp(320x320) @ XP[b](320x768) -> compact Xsp
    gemm(ASP, XP, S2, nullptr, NPAD, TC, NPAD, NPAD, TC, TC, N_,
         0, (long)NPAD * TC, (long)N_ * TC, B_, 1.0f);
    // x @ W_n
    gemm(x, W5, S3, nullptr, BNT, 64, 64, 64, 64, 64, BNT, 0, 0, 0, 1, 1.0f);
    ODEG_x0n<<<gr(BIG), 256, 0, stream>>>(x, S2, S3, ATTN, alpha_n, S4);

    // ---- MHSA (Q/K node-padded for guard-free score WMMA) ----
    gemm(XP, wq, S0, bq, NPAD, TC, TC, TC, TC, TC, NPAD,
         (long)NPAD * TC, 0, (long)NPAD * TC, B_, 1.0f);
    gemm(XP, wk, S1, bk, NPAD, TC, TC, TC, TC, TC, NPAD,
         (long)NPAD * TC, 0, (long)NPAD * TC, B_, 1.0f);
    gemm(x, wv, VB, bv, BN, 256, TC, TC, 256, 256, BN, 0, 0, 0, 1, 1.0f);
    ODEG_mhsa<<<dim3(20, 32, 16), 32, 0, stream>>>(S0, S1, VB, X1);

    // ---- edge branch (rank-collapsed) ----
    ODEG_em<<<gr(BNT), 256, 0, stream>>>(x, EM);
    ODEG_empad<<<gr(B_ * NPAD * 16), 256, 0, stream>>>(EM, EMP);
    ODEG_eoep<<<gr(BNT), 256, 0, stream>>>(EM, w3_12, w4_12, EO, EP);
    ODEG_esums<<<B_, 144, 0, stream>>>(EO, EP, SO, SPb, CR);
    ODEG_atte<<<gr((long)BN * 144), 256, 0, stream>>>(EO, EP, SO, SPb, CR, ATTE);
    ODEG_v1e<<<gr(BNT), 256, 0, stream>>>(EM, ATTE, V1E);
    ODEG_v1t<<<gr(B_ * 12 * 64), 256, 0, stream>>>(fce_w, V1E, V1T);
    ODEG_m2<<<gr(B_ * 144 * 64), 256, 0, stream>>>(fce_w, ATTE, M2);
    ODEG_hprep<<<gr(N_ * 64), 256, 0, stream>>>(alpha_e, fce_w, ASP, H, RS);
    // A_sp @ em[b] (padded 320x16) -> AEM (pitch 16)
    gemm(ASP, EMP, AEM, nullptr, NPAD, 16, NPAD, NPAD, 16, 16, N_,
         0, (long)NPAD * 16, (long)N_ * 16, B_, 1.0f);
    ODEG_q1<<<gr(B_ * 12 * 64), 256, 0, stream>>>(H, AEM, Q1);
    ODEG_q2<<<1, 64, 0, stream>>>(H, RS, Q2);

    // ---- block3: 4 groups x 3 Euler steps (X1S in S2) ----
    for (int g = 0; g < 4; ++g) {
        ODEG_zinit<<<gr((long)B_ * NPAD * 64), 256, 0, stream>>>(X1, Z0P, ZP, g);
        for (int step = 0; step < 3; ++step) {
            ODEG_sumsq<<<B_, 64, 0, stream>>>(ZP, SS);
            gemm(ASP, ZP, ZSPP, nullptr, NPAD, 64, NPAD, NPAD, 64, 64, NPAD,
                 0, (long)NPAD * 64, (long)NPAD * 64, B_, 1.0f);
            gemm(ZP, W5 + (1 + g) * 4096, ZWP, nullptr, B_ * NPAD, 64, 64, 64, 64, 64,
                 B_ * NPAD, 0, 0, 0, 1, 1.0f);
            ODEG_zupdate<<<gr((long)B_ * NPAD * 64), 256, 0, stream>>>(
                ZP, ZSPP, ZWP, Z0P, SS, alpha_3, w3_1, w4_1, S2, g, step);
        }
    }

    // ---- clip + output heads ----
    ODEG_clip<<<gr(BIG), 256, 0, stream>>>(S2, S4, clipv, S3);               // X1C
    gemm(S3, fc1_w, S0, fc1_b, BNT, 64, 64, 64, 64, 64, BNT, 0, 0, 0, 1, 1.0f); // XG
    gemm(S4, fc1_w, S1, fc1_b, BNT, 64, 64, 64, 64, 64, BNT, 0, 0, 0, 1, 1.0f); // XL
    ODEG_xe<<<gr(BIG), 256, 0, stream>>>(Q1, Q2, V1T, M2, EM, fce_b, S5);    // XE
    ODEG_zc<<<gr(BIG), 256, 0, stream>>>(S0, S1, S5, x, S6);                 // Zc
    gemm(x,  fc2_w, S3, fc2_b, BNT, 64, 64, 64, 64, 64, BNT, 0, 0, 0, 1, 1.0f); // RZ1
    gemm(S6, fc2_w, S2, fc2_b, BNT, 64, 64, 64, 64, 64, BNT, 0, 0, 0, 1, 1.0f); // RZ2
    ODEG_final<<<gr(BIG), 256, 0, stream>>>(S3, S2, out);
}